// GAT_42185168781721
// MI455X (gfx1250) — compile-verified
//
#include <hip/hip_runtime.h>

#define N_NODES 50000
#define E_EDGES 800000
#define IN_DIM  256
#define HID     64
#define HEADS   4
#define OUT_DIM 32
#define NEG_SLOPE 0.2f

typedef float v2f __attribute__((ext_vector_type(2)));
typedef float v8f __attribute__((ext_vector_type(8)));

#define CDIV(a,b) (((a)+(b)-1)/(b))

// ---- order-preserving float<->uint mapping for atomic float max ----
__device__ __forceinline__ unsigned f2ord(float f) {
  unsigned u = __float_as_uint(f);
  return (u & 0x80000000u) ? ~u : (u | 0x80000000u);
}
__device__ __forceinline__ float ord2f(unsigned u) {
  return (u & 0x80000000u) ? __uint_as_float(u & 0x7fffffffu)
                           : __uint_as_float(~u);
}

// LDS byte offset of a __shared__ object (low 32 bits of the generic pointer
// hold the LDS aperture offset on AMDGPU).
__device__ __forceinline__ unsigned lds_off(const void* p) {
  return (unsigned)(unsigned long long)p;
}

// =====================================================================
// FP32 WMMA GEMM: C[M,Nc] = A[M,K] * B[K,Nc]  (row-major, K % 64 == 0)
// 128 threads = 4 waves; block tile 128x64.
// wave w -> M-tiles {2w, 2w+1}; 4 column sub-tiles; 8 accumulators/wave.
// A tile staged with CDNA5 async global->LDS (global_load_async_to_lds_b128,
// drained by s_wait_asynccnt). B staged in k-pair-interleaved layout so a
// B fragment is a single aligned ds_load_b64 (no repack movs).
// =====================================================================
__global__ __launch_bounds__(128)
void wmma_gemm_kernel(const float* __restrict__ A, const float* __restrict__ B,
                      float* __restrict__ C, int M, int K, int Nc) {
  __shared__ float sA[128][68];    // pitch 68: conflict-free A-frag b64 reads
  __shared__ float sBp[32][160];   // [k/2][2*c + (k&1)]; pitch 160 (=32 mod 64)
  const int tid  = threadIdx.x;
  const int wave = tid >> 5;
  const int lane = tid & 31;
  const int rowBase = blockIdx.x * 128;
  const int colBase = blockIdx.y * 64;

  v8f acc[2][4] = {};

  const int n  = lane & 15;        // M (A-frag) / N (B-frag) within 16-tile
  const int ko = (lane >> 4) * 2;  // half-wave carries K={0,1} or K={2,3}

  for (int k0 = 0; k0 < K; k0 += 64) {
    // ---- A tile 128x16 (k cols 0..15): async copy, 16B per op, 4 ops/thread ----
    for (int i = tid; i < 128 * 4; i += 128) {
      int r   = i >> 2;
      int seg = (i & 3) * 4;             // float offset within row (16B chunks)
      int gr  = rowBase + r;
      if (gr < M) {                      // skipped rows: stale LDS only feeds
        unsigned ld = lds_off(&sA[r][seg]);  // C rows that are never stored
        const float* gp = A + (size_t)gr * K + k0 + seg;
        asm volatile("global_load_async_to_lds_b128 %0, %1, off"
                     :: "v"(ld), "v"(gp) : "memory");
      }
    }
    // ---- B tile 64x64 -> pair-interleaved LDS (zero-fill OOB cols) ----
    for (int i = tid; i < 64 * 16; i += 128) {
      int r  = i >> 4;                   // k within chunk: 0..63
      int c4 = (i & 15) * 4;             // col: 0..60
      int gc = colBase + c4;
      float4 v = make_float4(0.f, 0.f, 0.f, 0.f);
      if (gc + 3 < Nc) v = *reinterpret_cast<const float4*>(B + (size_t)(k0 + r) * Nc + gc);
      int kp = r >> 1, par = r & 1;
      sBp[kp][2 * (c4 + 0) + par] = v.x;
      sBp[kp][2 * (c4 + 1) + par] = v.y;
      sBp[kp][2 * (c4 + 2) + par] = v.z;
      sBp[kp][2 * (c4 + 3) + par] = v.w;
    }
    asm volatile("s_wait_asynccnt 0x0" ::: "memory");
    __syncthreads();

    #pragma unroll
    for (int kk = 0; kk < 64; kk += 4) {
      v2f a0 = *reinterpret_cast<const v2f*>(&sA[wave * 32 + n][kk + ko]);
      v2f a1 = *reinterpret_cast<const v2f*>(&sA[wave * 32 + 16 + n][kk + ko]);
      const int kp = (kk + ko) >> 1;
      #pragma unroll
      for (int nt = 0; nt < 4; ++nt) {
        v2f b = *reinterpret_cast<const v2f*>(&sBp[kp][2 * (nt * 16 + n)]);
        acc[0][nt] = __builtin_amdgcn_wmma_f32_16x16x4_f32(
            false, a0, false, b, (short)0, acc[0][nt], false, false);
        acc[1][nt] = __builtin_amdgcn_wmma_f32_16x16x4_f32(
            false, a1, false, b, (short)0, acc[1][nt], false, false);
      }
    }
    __syncthreads();
  }

  // C/D layout: VGPR r -> M = r (lanes 0-15) / r+8 (lanes 16-31), N = lane&15
  const int rOff = (lane >> 4) * 8;
  #pragma unroll
  for (int mt = 0; mt < 2; ++mt) {
    #pragma unroll
    for (int nt = 0; nt < 4; ++nt) {
      #pragma unroll
      for (int r = 0; r < 8; ++r) {
        int row = rowBase + wave * 32 + mt * 16 + r + rOff;
        int col = colBase + nt * 16 + n;
        if (row < M && col < Nc) C[(size_t)row * Nc + col] = acc[mt][nt][r];
      }
    }
  }
}

// =====================================================================
// Per-node attention scalars + init of max / denom accumulators
// =====================================================================
template<int H, int C>
__global__ void alpha_kernel(const float* __restrict__ h,
                             const float* __restrict__ a_src,
                             const float* __restrict__ a_dst,
                             float* __restrict__ as, float* __restrict__ ad,
                             unsigned* __restrict__ mord, float* __restrict__ den) {
  int t = blockIdx.x * blockDim.x + threadIdx.x;
  if (t >= N_NODES * H) return;
  int nn = t / H, hh = t % H;
  const float* hp = h + (size_t)nn * (H * C) + hh * C;
  float s0 = 0.f, s1 = 0.f;
  #pragma unroll 4
  for (int c = 0; c < C; ++c) {
    float hv = hp[c];
    s0 += hv * a_src[hh * C + c];
    s1 += hv * a_dst[hh * C + c];
  }
  as[t]   = s0;
  ad[t]   = s1;
  mord[t] = f2ord(-__builtin_huge_valf());
  den[t]  = 0.f;
}

__device__ __forceinline__ void edge_nodes(const int* src, const int* dst,
                                           long long eid, int nE, int& s, int& d) {
  if (eid < nE) { s = src[eid]; d = dst[eid]; }
  else          { s = d = (int)(eid - nE); }           // self-loop
}

// pass 1: segment max of leaky-relu(logit) over incoming edges
template<int H>
__global__ void edge_max_kernel(const int* __restrict__ src, const int* __restrict__ dst,
                                const float* __restrict__ as, const float* __restrict__ ad,
                                unsigned* __restrict__ mord) {
  long long t = (long long)blockIdx.x * blockDim.x + threadIdx.x;
  long long total = (long long)(E_EDGES + N_NODES) * H;
  if (t >= total) return;
  long long eid = t / H;
  int hh = (int)(t - eid * H);
  int s, d; edge_nodes(src, dst, eid, E_EDGES, s, d);
  float e = as[(size_t)s * H + hh] + ad[(size_t)d * H + hh];
  e = (e >= 0.f) ? e : NEG_SLOPE * e;
  atomicMax(&mord[(size_t)d * H + hh], f2ord(e));
}

// pass 2: segment sum of exp(e - max)
template<int H>
__global__ void edge_sum_kernel(const int* __restrict__ src, const int* __restrict__ dst,
                                const float* __restrict__ as, const float* __restrict__ ad,
                                const unsigned* __restrict__ mord, float* __restrict__ den) {
  long long t = (long long)blockIdx.x * blockDim.x + threadIdx.x;
  long long total = (long long)(E_EDGES + N_NODES) * H;
  if (t >= total) return;
  long long eid = t / H;
  int hh = (int)(t - eid * H);
  int s, d; edge_nodes(src, dst, eid, E_EDGES, s, d);
  float e = as[(size_t)s * H + hh] + ad[(size_t)d * H + hh];
  e = (e >= 0.f) ? e : NEG_SLOPE * e;
  float ex = __expf(e - ord2f(mord[(size_t)d * H + hh]));
  atomicAdd(&den[(size_t)d * H + hh], ex);
}

// pass 3: weighted scatter  agg[dst] += alpha * h[src]
// H*C/4 threads per edge, each handles one float4 of the feature vector.
template<int H, int C>
__global__ void edge_msg_kernel(const int* __restrict__ src, const int* __restrict__ dst,
                                const float* __restrict__ h,
                                const float* __restrict__ as, const float* __restrict__ ad,
                                const unsigned* __restrict__ mord,
                                const float* __restrict__ den,
                                float* __restrict__ agg) {
  const int TPE = H * C / 4;
  long long t = (long long)blockIdx.x * blockDim.x + threadIdx.x;
  long long eid = t / TPE;
  if (eid >= (long long)(E_EDGES + N_NODES)) return;
  int j = (int)(t - eid * TPE);
  int s, d; edge_nodes(src, dst, eid, E_EDGES, s, d);
  int hh = j / (C / 4);
  int c4 = (j % (C / 4)) * 4;
  float e = as[(size_t)s * H + hh] + ad[(size_t)d * H + hh];
  e = (e >= 0.f) ? e : NEG_SLOPE * e;
  float alpha = __expf(e - ord2f(mord[(size_t)d * H + hh])) /
                (den[(size_t)d * H + hh] + 1e-16f);
  const float4 hv = *reinterpret_cast<const float4*>(h + (size_t)s * (H * C) + hh * C + c4);
  float* op = agg + (size_t)d * (H * C) + hh * C + c4;
  atomicAdd(op + 0, hv.x * alpha);
  atomicAdd(op + 1, hv.y * alpha);
  atomicAdd(op + 2, hv.z * alpha);
  atomicAdd(op + 3, hv.w * alpha);
}

// =====================================================================
// elementwise: zero / bias(+optional ELU)
// =====================================================================
__global__ void zero_kernel(float* __restrict__ p, int n) {
  int t = blockIdx.x * blockDim.x + threadIdx.x;
  if (t < n) p[t] = 0.f;
}

template<bool ELU>
__global__ void bias_act_kernel(float* __restrict__ p, const float* __restrict__ bias,
                                int n, int ldb) {
  int t = blockIdx.x * blockDim.x + threadIdx.x;
  if (t >= n) return;
  float v = p[t] + bias[t % ldb];
  if (ELU) v = (v > 0.f) ? v : expm1f(v);
  p[t] = v;
}

// =====================================================================
// launch
// =====================================================================
extern "C" void kernel_launch(void* const* d_in, const int* in_sizes, int n_in,
                              void* d_out, int out_size, void* d_ws, size_t ws_size,
                              hipStream_t stream) {
  const float* x    = (const float*)d_in[0];
  const int*   ei   = (const int*)  d_in[1];
  const float* W1   = (const float*)d_in[2];
  const float* a_s1 = (const float*)d_in[3];
  const float* a_d1 = (const float*)d_in[4];
  const float* b1   = (const float*)d_in[5];
  const float* W2   = (const float*)d_in[6];
  const float* a_s2 = (const float*)d_in[7];
  const float* a_d2 = (const float*)d_in[8];
  const float* b2   = (const float*)d_in[9];
  float* out = (float*)d_out;

  const int* src = ei;             // edge_index row 0
  const int* dst = ei + E_EDGES;   // edge_index row 1

  // workspace layout (~113 MB of f32; fits global L2)
  float*    h1   = (float*)d_ws;                               // N*256
  float*    agg1 = h1   + (size_t)N_NODES * 256;               // N*256
  float*    as1  = agg1 + (size_t)N_NODES * 256;               // N*4
  float*    ad1  = as1  + (size_t)N_NODES * HEADS;             // N*4
  unsigned* m1   = (unsigned*)(ad1 + (size_t)N_NODES * HEADS); // N*4
  float*    den1 = (float*)(m1 + (size_t)N_NODES * HEADS);     // N*4
  float*    h2   = den1 + (size_t)N_NODES * HEADS;             // N*32
  float*    as2  = h2   + (size_t)N_NODES * OUT_DIM;           // N
  float*    ad2  = as2  + N_NODES;                             // N
  unsigned* m2   = (unsigned*)(ad2 + N_NODES);                 // N
  float*    den2 = (float*)(m2 + N_NODES);                     // N

  const int BT = 256;
  const long long EP = (long long)E_EDGES + N_NODES;           // with self loops

  // ---------------- layer 1 ----------------
  zero_kernel<<<CDIV(N_NODES * 256, BT), BT, 0, stream>>>(agg1, N_NODES * 256);

  dim3 g1(CDIV(N_NODES, 128), IN_DIM / 64);                    // 391 x 4
  wmma_gemm_kernel<<<g1, 128, 0, stream>>>(x, W1, h1, N_NODES, IN_DIM, HEADS * HID);

  alpha_kernel<HEADS, HID><<<CDIV(N_NODES * HEADS, BT), BT, 0, stream>>>(
      h1, a_s1, a_d1, as1, ad1, m1, den1);

  edge_max_kernel<HEADS><<<(int)CDIV(EP * HEADS, BT), BT, 0, stream>>>(
      src, dst, as1, ad1, m1);
  edge_sum_kernel<HEADS><<<(int)CDIV(EP * HEADS, BT), BT, 0, stream>>>(
      src, dst, as1, ad1, m1, den1);
  edge_msg_kernel<HEADS, HID><<<(int)CDIV(EP * (HEADS * HID / 4), BT), BT, 0, stream>>>(
      src, dst, h1, as1, ad1, m1, den1, agg1);

  bias_act_kernel<true><<<CDIV(N_NODES * 256, BT), BT, 0, stream>>>(
      agg1, b1, N_NODES * 256, HEADS * HID);                   // +b1, ELU

  // ---------------- layer 2 ----------------
  dim3 g2(CDIV(N_NODES, 128), 1);
  wmma_gemm_kernel<<<g2, 128, 0, stream>>>(agg1, W2, h2, N_NODES, HEADS * HID, OUT_DIM);

  alpha_kernel<1, OUT_DIM><<<CDIV(N_NODES, BT), BT, 0, stream>>>(
      h2, a_s2, a_d2, as2, ad2, m2, den2);

  zero_kernel<<<CDIV(N_NODES * OUT_DIM, BT), BT, 0, stream>>>(out, N_NODES * OUT_DIM);

  edge_max_kernel<1><<<(int)CDIV(EP, BT), BT, 0, stream>>>(src, dst, as2, ad2, m2);
  edge_sum_kernel<1><<<(int)CDIV(EP, BT), BT, 0, stream>>>(src, dst, as2, ad2, m2, den2);
  edge_msg_kernel<1, OUT_DIM><<<(int)CDIV(EP * (OUT_DIM / 4), BT), BT, 0, stream>>>(
      src, dst, h2, as2, ad2, m2, den2, out);

  bias_act_kernel<false><<<CDIV(N_NODES * OUT_DIM, BT), BT, 0, stream>>>(
      out, b2, N_NODES * OUT_DIM, OUT_DIM);
}